// ReprojectionLayer_21766894256505
// MI455X (gfx1250) — compile-verified
//
#include <hip/hip_runtime.h>
#include <stdint.h>

// Problem constants from reference setup_inputs():
//   heatmaps [B=2][C=12][J=23][H=128][W=128] f32
//   lookup   [C][160][160][160] i32, ROI cube G=64, half=32
//   out      [B][J][64][64][64] f32
#define B_    2
#define C_    12
#define J_    23
#define HW_   (128 * 128)
#define GF_   160
#define G_    64
#define HALF_ 32
#define NTHREADS 256

typedef __attribute__((address_space(1))) int as1_int;
typedef __attribute__((address_space(3))) int as3_int;

// CDNA5 async global->LDS copy (ASYNCcnt path). Builtin when available,
// otherwise raw gfx1250 mnemonic — both emit global_load_async_to_lds_b32.
__device__ __forceinline__ void async_idx_load(const int* gp, int* lp) {
#if __has_builtin(__builtin_amdgcn_global_load_async_to_lds_b32)
  __builtin_amdgcn_global_load_async_to_lds_b32(
      (as1_int*)(int*)gp, (as3_int*)lp, /*offset=*/0, /*cpol=*/0);
#else
  unsigned lds_off = (unsigned)(uintptr_t)(as3_int*)lp;
  asm volatile("global_load_async_to_lds_b32 %0, %1, off"
               :: "v"(lds_off), "v"((unsigned long long)(uintptr_t)gp)
               : "memory");
#endif
}

__device__ __forceinline__ void wait_async_all() {
#if __has_builtin(__builtin_amdgcn_s_wait_asynccnt)
  __builtin_amdgcn_s_wait_asynccnt(0);
#else
  asm volatile("s_wait_asynccnt 0x0" ::: "memory");
#endif
}

__global__ __launch_bounds__(NTHREADS)
void reprojection_gather_mean_kernel(
    const float* __restrict__ hm,       // [B][C][J][HW]
    const float* __restrict__ center,   // [B][3]
    const int*   __restrict__ lookup,   // [C][GF][GF][GF]
    const float* __restrict__ offs,     // [3]
    const float* __restrict__ spacing,  // [1]
    float* __restrict__ out)            // [B][J][G][G][G]
{
  // Index rows staged as [z][c] so each thread reads 12 contiguous ints.
  __shared__ __align__(16) int s_idx[G_ * C_];   // 3 KB

  const int x   = blockIdx.x;
  const int y   = blockIdx.y;
  const int b   = blockIdx.z;
  const int tid = threadIdx.x;

  // center_indices = ((center - offset) / spacing).astype(int32)  (trunc)
  const float spv = spacing[0];
  const int ci0 = (int)((center[b * 3 + 0] - offs[0]) / spv);
  const int ci1 = (int)((center[b * 3 + 1] - offs[1]) / spv);
  const int ci2 = (int)((center[b * 3 + 2] - offs[2]) / spv);
  const int cx = ci0 - HALF_ + x;
  const int cy = ci1 - HALF_ + y;
  const int cz = ci2 - HALF_;

  // Stage 12 camera index rows (64 contiguous ints each) into LDS.
  // 768 elements over 256 threads == exactly 3 full-EXEC async issues;
  // straight-line (no loop/branch scaffolding).
  {
    const size_t xy = (size_t)cx * (GF_ * GF_) + (size_t)cy * GF_ + (size_t)cz;
#pragma unroll
    for (int k = 0; k < 3; ++k) {
      const int i = tid + k * NTHREADS;     // 0..767, uniform trip count
      const int c = i >> 6;                 // camera 0..11
      const int z = i & 63;                 // voxel z 0..63
      const int* gp = lookup + (size_t)c * (GF_ * GF_ * GF_) + xy + (size_t)z;
      async_idx_load(gp, &s_idx[z * C_ + c]);
    }
  }
  wait_async_all();
  __syncthreads();

  const int z  = tid & 63;        // this thread's z voxel
  const int jg = tid >> 6;        // joint group 0..3 (stride 4 over joints)

  // 12 indices for this z: three ds_load_b128 ((z*12*4) is 16B-aligned).
  int idx[C_];
  const int4* r4 = reinterpret_cast<const int4*>(&s_idx[z * C_]);
  const int4 q0 = r4[0], q1 = r4[1], q2 = r4[2];
  idx[0] = q0.x; idx[1]  = q0.y; idx[2]  = q0.z; idx[3]  = q0.w;
  idx[4] = q1.x; idx[5]  = q1.y; idx[6]  = q1.z; idx[7]  = q1.w;
  idx[8] = q2.x; idx[9]  = q2.y; idx[10] = q2.z; idx[11] = q2.w;

  const float* hb = hm + (size_t)b * C_ * J_ * HW_;
  const size_t ob = ((size_t)b * J_ * G_ + (size_t)x) * (G_ * G_) +
                    (size_t)y * G_ + (size_t)z;   // out index for j=0

  // Process joints in pairs (j, j+4): 24 gathers in flight per iteration,
  // two independent accumulators -> better L2-gather latency hiding.
  int j = jg;
  for (; j + 4 < J_; j += 8) {
    const float* p0 = hb + (size_t)j * HW_;
    const float* p1 = hb + (size_t)(j + 4) * HW_;
    float a0 = 0.0f, a1 = 0.0f;
#pragma unroll
    for (int c = 0; c < C_; ++c) {
      const size_t co = (size_t)c * (J_ * HW_) + (size_t)idx[c];
      a0 += p0[co];
      a1 += p1[co];
    }
    __builtin_nontemporal_store(a0 * (1.0f / 12.0f),
                                out + ob + (size_t)j * (G_ * G_ * G_));
    __builtin_nontemporal_store(a1 * (1.0f / 12.0f),
                                out + ob + (size_t)(j + 4) * (G_ * G_ * G_));
  }
  if (j < J_) {                   // tail (only jg==3 hits this: j==19)
    const float* p0 = hb + (size_t)j * HW_;
    float a0 = 0.0f;
#pragma unroll
    for (int c = 0; c < C_; ++c)
      a0 += p0[(size_t)c * (J_ * HW_) + (size_t)idx[c]];
    __builtin_nontemporal_store(a0 * (1.0f / 12.0f),
                                out + ob + (size_t)j * (G_ * G_ * G_));
  }
}

extern "C" void kernel_launch(void* const* d_in, const int* in_sizes, int n_in,
                              void* d_out, int out_size, void* d_ws, size_t ws_size,
                              hipStream_t stream) {
  const float* hm      = (const float*)d_in[0];
  const float* center  = (const float*)d_in[1];
  const int*   lookup  = (const int*)d_in[2];
  const float* offs    = (const float*)d_in[3];
  const float* spacing = (const float*)d_in[4];
  float* out = (float*)d_out;

  dim3 grid(G_, G_, B_);   // (x, y, batch) -> 8192 blocks
  reprojection_gather_mean_kernel<<<grid, NTHREADS, 0, stream>>>(
      hm, center, lookup, offs, spacing, out);
}